// GCN_56659208568910
// MI455X (gfx1250) — compile-verified
//
#include <hip/hip_runtime.h>

typedef __attribute__((ext_vector_type(2))) float v2f;
typedef __attribute__((ext_vector_type(8))) float v8f;

#define N_NODES 50000
#define FDIM    128
#define XS_STRIDE 132   // 132 % 64 banks -> lane l hits bank (4*l)%64, conflict-free

// ---------------- degree / norm precompute ----------------

__global__ void deg_init_kernel(float* __restrict__ deg) {
    int n = blockIdx.x * 256 + threadIdx.x;
    if (n < N_NODES) deg[n] = 1.0f;               // self loop contributes 1
}

__global__ void deg_edges_kernel(const int* __restrict__ dst, int E,
                                 float* __restrict__ deg) {
    int e = blockIdx.x * 256 + threadIdx.x;
    if (e < E) atomicAdd(&deg[dst[e]], 1.0f);
}

__global__ void dinv_kernel(const float* __restrict__ deg,
                            float* __restrict__ dinv) {
    int n = blockIdx.x * 256 + threadIdx.x;
    if (n < N_NODES) dinv[n] = rsqrtf(deg[n]);    // deg >= 1 always
}

__global__ void edge_norm_kernel(const int* __restrict__ src,
                                 const int* __restrict__ dst,
                                 const float* __restrict__ dinv, int E,
                                 float* __restrict__ normE) {
    int e = blockIdx.x * 256 + threadIdx.x;
    if (e < E) normE[e] = dinv[src[e]] * dinv[dst[e]];
}

// ---------------- GEMM: H = (relu?)X @ W, X:[N,128], W:[128,128] ----------------
// 1 block = 8 waves; wave w computes the 16x16 tile at (blockIdx.x*16, w*16).
// V_WMMA_F32_16X16X4_F32, K swept 0..124 in steps of 4.

__global__ __launch_bounds__(256) void gemm_wmma_kernel(
        const float* __restrict__ X, const float* __restrict__ W,
        float* __restrict__ H, int relu_in) {
    __shared__ float xs[16 * XS_STRIDE];

    const int tid = threadIdx.x;
    const int m0  = blockIdx.x * 16;

    // Stage 16x128 X tile into LDS as float4 (fused input ReLU).
    for (int i = tid; i < 16 * 32; i += 256) {
        int r  = i >> 5;
        int c4 = i & 31;
        float4 v = *((const float4*)(X + (size_t)(m0 + r) * FDIM) + c4);
        if (relu_in) {
            v.x = fmaxf(v.x, 0.0f); v.y = fmaxf(v.y, 0.0f);
            v.z = fmaxf(v.z, 0.0f); v.w = fmaxf(v.w, 0.0f);
        }
        *(float4*)(&xs[r * XS_STRIDE + c4 * 4]) = v;
    }
    __syncthreads();

    const int lane = tid & 31;
    const int wave = tid >> 5;
    const int n0   = wave * 16;
    const int lm   = lane & 15;           // M (for A) / N (for B) within tile
    const int khi  = (lane >> 4) << 1;    // lanes 16-31 hold K+2, K+3

    v8f acc = {};
    #pragma unroll
    for (int k0 = 0; k0 < FDIM; k0 += 4) {
        const int kk = k0 + khi;
        v2f a, b;
        // A 16x4 f32 layout: lane lm row, VGPR0=K kk, VGPR1=K kk+1
        a.x = xs[lm * XS_STRIDE + kk];
        a.y = xs[lm * XS_STRIDE + kk + 1];
        // B 4x16 f32 layout: row kk (VGPR0) / kk+1 (VGPR1), col n0+lm
        b.x = W[(size_t)kk * FDIM + n0 + lm];
        b.y = W[(size_t)(kk + 1) * FDIM + n0 + lm];
        acc = __builtin_amdgcn_wmma_f32_16x16x4_f32(
                  false, a, false, b, (short)0, acc, false, false);
    }

    // C/D layout: VGPR v -> M = v + (lane<16 ? 0 : 8), N = lane&15
    const int mbase = m0 + ((lane >> 4) << 3);
    #pragma unroll
    for (int v = 0; v < 8; ++v) {
        H[(size_t)(mbase + v) * FDIM + n0 + lm] = acc[v];
    }
}

// ---------------- aggregation ----------------
// out[n,f] = b[f] + dinv[n]^2 * H[n,f]   (bias + self-loop term), float4.

__global__ void agg_init_kernel(const float* __restrict__ H,
                                const float* __restrict__ dinv,
                                const float* __restrict__ bias,
                                float* __restrict__ out) {
    int idx = blockIdx.x * 256 + threadIdx.x;       // over N*F/4
    if (idx >= N_NODES * (FDIM / 4)) return;
    int n  = idx >> 5;
    int f4 = idx & 31;
    float s = dinv[n]; s *= s;
    float4 h = ((const float4*)H)[idx];
    float4 b = ((const float4*)bias)[f4];
    float4 o;
    o.x = b.x + s * h.x; o.y = b.y + s * h.y;
    o.z = b.z + s * h.z; o.w = b.w + s * h.w;
    ((float4*)out)[idx] = o;
}

// One thread per (edge, feature): coalesced gather + scatter-add.
__global__ void agg_edges_kernel(const float* __restrict__ H,
                                 const int* __restrict__ src,
                                 const int* __restrict__ dst,
                                 const float* __restrict__ normE,
                                 int E, float* __restrict__ out) {
    unsigned idx = blockIdx.x * 256u + threadIdx.x;
    unsigned e = idx >> 7;
    if (e >= (unsigned)E) return;
    int f = idx & 127;
    int s = src[e];
    int d = dst[e];
    float m = normE[e] * H[(size_t)s * FDIM + f];
    atomicAdd(&out[(size_t)d * FDIM + f], m);
}

__global__ void dup_kernel(const float* __restrict__ in, float* __restrict__ out) {
    int idx = blockIdx.x * 256 + threadIdx.x;       // over N*F/4
    if (idx < N_NODES * (FDIM / 4))
        ((float4*)out)[idx] = ((const float4*)in)[idx];
}

// ---------------- driver ----------------

extern "C" void kernel_launch(void* const* d_in, const int* in_sizes, int n_in,
                              void* d_out, int out_size, void* d_ws, size_t ws_size,
                              hipStream_t stream) {
    const float* x  = (const float*)d_in[0];
    const int*   ei = (const int*)d_in[1];
    const float* W1 = (const float*)d_in[2];
    const float* b1 = (const float*)d_in[3];
    const float* W2 = (const float*)d_in[4];
    const float* b2 = (const float*)d_in[5];
    const float* W3 = (const float*)d_in[6];
    const float* b3 = (const float*)d_in[7];

    const int E = in_sizes[1] / 2;
    const int* srcv = ei;          // edge_index[0]
    const int* dstv = ei + E;      // edge_index[1]

    float* ws    = (float*)d_ws;
    float* deg   = ws;
    float* dinv  = deg + N_NODES;
    float* normE = dinv + N_NODES;
    float* Hbuf  = normE + ((E + 3) & ~3);
    float* Abuf  = Hbuf + (size_t)N_NODES * FDIM;

    float* out1 = (float*)d_out;
    float* out2 = out1 + (size_t)N_NODES * FDIM;

    const int nodeBlocks = (N_NODES + 255) / 256;
    const int edgeBlocks = (E + 255) / 256;
    const int nf4Blocks  = (N_NODES * (FDIM / 4) + 255) / 256;
    const int efBlocks   = (int)(((size_t)E * FDIM + 255) / 256);
    const int gemmBlocks = N_NODES / 16;   // 3125

    // Normalization coefficients (once, shared by all layers).
    deg_init_kernel<<<nodeBlocks, 256, 0, stream>>>(deg);
    deg_edges_kernel<<<edgeBlocks, 256, 0, stream>>>(dstv, E, deg);
    dinv_kernel<<<nodeBlocks, 256, 0, stream>>>(deg, dinv);
    edge_norm_kernel<<<edgeBlocks, 256, 0, stream>>>(srcv, dstv, dinv, E, normE);

    // Layer 1: h1 = agg(x @ W1) + b1   (ReLU deferred into layer-2 GEMM load)
    gemm_wmma_kernel<<<gemmBlocks, 256, 0, stream>>>(x, W1, Hbuf, 0);
    agg_init_kernel<<<nf4Blocks, 256, 0, stream>>>(Hbuf, dinv, b1, Abuf);
    agg_edges_kernel<<<efBlocks, 256, 0, stream>>>(Hbuf, srcv, dstv, normE, E, Abuf);

    // Layer 2: h2 = agg(relu(h1) @ W2) + b2
    gemm_wmma_kernel<<<gemmBlocks, 256, 0, stream>>>(Abuf, W2, Hbuf, 1);
    agg_init_kernel<<<nf4Blocks, 256, 0, stream>>>(Hbuf, dinv, b2, Abuf);
    agg_edges_kernel<<<efBlocks, 256, 0, stream>>>(Hbuf, srcv, dstv, normE, E, Abuf);

    // Layer 3: out = agg(relu(h2) @ W3) + b3   (no final ReLU)
    gemm_wmma_kernel<<<gemmBlocks, 256, 0, stream>>>(Abuf, W3, Hbuf, 1);
    agg_init_kernel<<<nf4Blocks, 256, 0, stream>>>(Hbuf, dinv, b3, out1);
    agg_edges_kernel<<<efBlocks, 256, 0, stream>>>(Hbuf, srcv, dstv, normE, E, out1);

    // Reference returns (h, h): duplicate into second half of d_out.
    dup_kernel<<<nf4Blocks, 256, 0, stream>>>(out1, out2);
}